// GCN_67473936220739
// MI455X (gfx1250) — compile-verified
//
#include <hip/hip_runtime.h>
#include <hip/hip_bf16.h>
#include <math.h>

typedef __attribute__((ext_vector_type(2))) float v2f;
typedef __attribute__((ext_vector_type(8))) float v8f;
typedef int v4i __attribute__((vector_size(4 * sizeof(int))));

#define NNODE 1536
#define HID 64

// ---------------------------------------------------------------------------
// Async global -> LDS helpers (gfx1250 GLOBAL_LOAD_ASYNC_TO_LDS_*)
// Builtin signature (from compiler diagnostics):
//   b128: (v4i* src, __shared__ v4i* lds, int offset, int cpol)
//   b32 : (int* src, __shared__ int*  lds, int offset, int cpol)
// ---------------------------------------------------------------------------
__device__ __forceinline__ void async_g2l_b128(const float* g, float* l) {
  __builtin_amdgcn_global_load_async_to_lds_b128(
      (v4i*)g,
      (__attribute__((address_space(3))) v4i*)l, 0, 0);
}
__device__ __forceinline__ void async_g2l_b32(const float* g, float* l) {
  __builtin_amdgcn_global_load_async_to_lds_b32(
      (int*)g,
      (__attribute__((address_space(3))) int*)l, 0, 0);
}
__device__ __forceinline__ void wait_async0() {
  asm volatile("s_wait_asynccnt 0x0" ::: "memory");
}

// ---------------------------------------------------------------------------
// deg[row] = sum_j adj[row, j]
// ---------------------------------------------------------------------------
__global__ __launch_bounds__(256) void rowsum_kernel(const float* __restrict__ adj,
                                                     float* __restrict__ deg, int n) {
  __shared__ float red[8];
  const int row = blockIdx.x;
  const float* p = adj + (size_t)row * n;
  float s = 0.f;
  for (int j = threadIdx.x; j < n; j += 256) s += p[j];
#pragma unroll
  for (int off = 16; off > 0; off >>= 1) s += __shfl_xor(s, off, 32);
  if ((threadIdx.x & 31) == 0) red[threadIdx.x >> 5] = s;
  __syncthreads();
  if (threadIdx.x == 0) {
    float t = 0.f;
#pragma unroll
    for (int i = 0; i < 8; ++i) t += red[i];
    deg[row] = t;
  }
}

// ---------------------------------------------------------------------------
// WMMA fp32 partial GEMM:
//   Cpart[seg][M x 64] = A[M x (k-range of seg)] @ B[(k-range) x 64]
// gridDim.y = #K segments (seg length = Kseg, multiple of 32).
// Block = 128 threads (4 wave32), tile = 64 rows x 64 cols.
// Double-buffered LDS, async global->LDS staging; B staged TRANSPOSED so both
// fragments are single b64 LDS loads with conflict-free stride-36 banking.
// ---------------------------------------------------------------------------
__global__ __launch_bounds__(128) void wmma_gemm_part(
    const float* __restrict__ A, const float* __restrict__ B,
    float* __restrict__ Cpart, int M, int K, int Kseg) {
  constexpr int MB = 64, KB = 32;
  constexpr int LDT = KB + 4;  // 36 floats: 16B-aligned rows, conflict-free b64 reads
  __shared__ float sA[2][MB * LDT];    // row-major A tile      (2 x 9216 B)
  __shared__ float sBt[2][HID * LDT];  // transposed B tile [col][k] (2 x 9216 B)

  const int tid = threadIdx.x;
  const int wave = tid >> 5;
  const int lane = tid & 31;
  const int l15 = lane & 15;
  const int khalf = lane >> 4;
  const int mBase = blockIdx.x * MB;
  const int mWave = wave * 16;
  const int seg = blockIdx.y;
  const int k0 = seg * Kseg;

  v8f acc[4];
#pragma unroll
  for (int i = 0; i < 4; ++i)
#pragma unroll
    for (int r = 0; r < 8; ++r) acc[i][r] = 0.f;

  auto stage = [&](int kc, int bf) {
    // A tile: 64 x 32 floats = 512 float4, 4 async b128 per thread.
#pragma unroll
    for (int c = 0; c < 4; ++c) {
      const int idx = tid + c * 128;
      const int row = idx >> 3;
      const int col4 = (idx & 7) << 2;
      async_g2l_b128(A + (size_t)(mBase + row) * K + kc + col4,
                     &sA[bf][row * LDT + col4]);
    }
    // B tile transposed: 32 x 64 floats, 16 async b32 per thread
    // (coalesced global reads; scattered LDS writes handled by the async unit).
#pragma unroll
    for (int c = 0; c < 16; ++c) {
      const int idx = tid + c * 128;
      const int row = idx >> 6;   // k within chunk
      const int col = idx & 63;   // n
      async_g2l_b32(B + (size_t)(kc + row) * HID + col,
                    &sBt[bf][col * LDT + row]);
    }
  };

  stage(k0, 0);
  wait_async0();
  __syncthreads();

  int cur = 0;
  for (int kc = k0; kc < k0 + Kseg; kc += KB) {
    if (kc + KB < k0 + Kseg) stage(kc + KB, cur ^ 1);  // prefetch next chunk

    const float* a = &sA[cur][0];
    const float* bt = &sBt[cur][0];
#pragma unroll
    for (int kk = 0; kk < KB; kk += 4) {
      // A fragment (16x4 f32): v0/v1 = K {0,1} (lanes 0-15) or {2,3} (16-31)
      v2f af = *(const v2f*)(a + (mWave + l15) * LDT + kk + 2 * khalf);
#pragma unroll
      for (int nt = 0; nt < 4; ++nt) {
        // B fragment (4x16 f32) from transposed tile: single b64 load
        v2f bf2 = *(const v2f*)(bt + (nt * 16 + l15) * LDT + kk + 2 * khalf);
        acc[nt] = __builtin_amdgcn_wmma_f32_16x16x4_f32(
            false, af, false, bf2, (short)0, acc[nt], false, false);
      }
    }
    wait_async0();
    __syncthreads();
    cur ^= 1;
  }

  // C/D layout: VGPR r -> M = r + 8*khalf, N = l15 (+16*nt)
  float* Cp = Cpart + (size_t)seg * M * HID;
#pragma unroll
  for (int nt = 0; nt < 4; ++nt) {
#pragma unroll
    for (int r = 0; r < 8; ++r) {
      const int m = mBase + mWave + r + 8 * khalf;
      Cp[(size_t)m * HID + nt * 16 + l15] = acc[nt][r];
    }
  }
}

// ---------------------------------------------------------------------------
// Reduce K-split partials + epilogue.
// MODE 0: plain sum            MODE 1: relu(sum + bias[col])
// MODE 2: elu(deg*aux+sum+b)   MODE 3: deg*aux + sum + bias
// ---------------------------------------------------------------------------
template <int MODE>
__global__ __launch_bounds__(256) void reduce_epi_kernel(
    const float* __restrict__ P, int S, float* __restrict__ C, int M,
    const float* __restrict__ bias, const float* __restrict__ deg,
    const float* __restrict__ aux) {
  const int i = blockIdx.x * 256 + threadIdx.x;
  if (i >= M * HID) return;
  float s = 0.f;
  for (int sg = 0; sg < S; ++sg) s += P[(size_t)sg * M * HID + i];
  const int m = i >> 6;
  const int col = i & 63;
  if constexpr (MODE == 1) {
    s = fmaxf(s + bias[col], 0.f);
  } else if constexpr (MODE == 2) {
    float t = deg[m] * aux[i] + s + bias[col];
    s = t > 0.f ? t : expm1f(t);
  } else if constexpr (MODE == 3) {
    s = deg[m] * aux[i] + s + bias[col];
  }
  C[i] = s;
}

// ---------------------------------------------------------------------------
// Row-wise log_softmax over 64 columns; one wave32 per row (2 cols/lane).
// ---------------------------------------------------------------------------
__global__ __launch_bounds__(256) void logsoftmax_kernel(const float* __restrict__ X,
                                                         float* __restrict__ out, int M) {
  const int row = blockIdx.x * 8 + (threadIdx.x >> 5);
  if (row >= M) return;
  const int lane = threadIdx.x & 31;
  const float* x = X + (size_t)row * HID;
  float a = x[lane];
  float b = x[lane + 32];
  float m = fmaxf(a, b);
#pragma unroll
  for (int off = 16; off > 0; off >>= 1) m = fmaxf(m, __shfl_xor(m, off, 32));
  float s = expf(a - m) + expf(b - m);
#pragma unroll
  for (int off = 16; off > 0; off >>= 1) s += __shfl_xor(s, off, 32);
  const float lse = m + logf(s);
  out[(size_t)row * HID + lane] = a - lse;
  out[(size_t)row * HID + lane + 32] = b - lse;
}

// ---------------------------------------------------------------------------
// Launch: full GCN forward.
// inputs: 0 x_init, 1 adj, 2 adj1, 3 fully_connected_graph, 4 W1, 5 b1,
//         6 W2, 7 b2, 8 W3, 9 b3   (adj / fully_connected_graph unused)
// ---------------------------------------------------------------------------
extern "C" void kernel_launch(void* const* d_in, const int* in_sizes, int n_in,
                              void* d_out, int out_size, void* d_ws, size_t ws_size,
                              hipStream_t stream) {
  (void)in_sizes; (void)n_in; (void)out_size; (void)ws_size;
  const float* x_init = (const float*)d_in[0];
  const float* adj1   = (const float*)d_in[2];
  const float* W1 = (const float*)d_in[4];
  const float* b1 = (const float*)d_in[5];
  const float* W2 = (const float*)d_in[6];
  const float* b2 = (const float*)d_in[7];
  const float* W3 = (const float*)d_in[8];
  const float* b3 = (const float*)d_in[9];
  float* out = (float*)d_out;

  constexpr int NE = NNODE * HID;  // 98304 elements per [N x 64] matrix
  float* w    = (float*)d_ws;
  float* deg  = w;               // 1536 (padded to 2048)
  float* slab = w + 2048;        // 4 partial slabs: 4 * NE
  float* buf0 = slab + 4 * NE;   // T1 / A3
  float* buf1 = buf0 + NE;       // x1 / x3
  float* buf2 = buf1 + NE;       // A2
  float* buf3 = buf2 + NE;       // B2 / B3
  float* buf4 = buf3 + NE;       // x2

  const dim3 blk(128);
  const dim3 g1(NNODE / 64, 1);   // unsplit GEMM
  const dim3 g4(NNODE / 64, 4);   // K split 4 ways
  const dim3 rgrid((NE + 255) / 256);

  // deg = rowsum(adj1)
  rowsum_kernel<<<NNODE, 256, 0, stream>>>(adj1, deg, NNODE);

  // T1 = x_init @ W1   (K=512, split 4 x 128)
  wmma_gemm_part<<<g4, blk, 0, stream>>>(x_init, W1, slab, NNODE, 512, 128);
  reduce_epi_kernel<0><<<rgrid, 256, 0, stream>>>(slab, 4, buf0, NNODE,
                                                  nullptr, nullptr, nullptr);
  // x1 = relu(adj1 @ T1 + b1)   (K=1536, split 4 x 384)
  wmma_gemm_part<<<g4, blk, 0, stream>>>(adj1, buf0, slab, NNODE, NNODE, 384);
  reduce_epi_kernel<1><<<rgrid, 256, 0, stream>>>(slab, 4, buf1, NNODE,
                                                  b1, nullptr, nullptr);
  // A2 = x1 @ W2_top ; B2 = x1 @ W2_bot   (K=64, unsplit)
  wmma_gemm_part<<<g1, blk, 0, stream>>>(buf1, W2, buf2, NNODE, HID, HID);
  wmma_gemm_part<<<g1, blk, 0, stream>>>(buf1, W2 + HID * HID, buf3, NNODE, HID, HID);
  // x2 = elu(deg*A2 + adj1 @ B2 + b2)
  wmma_gemm_part<<<g4, blk, 0, stream>>>(adj1, buf3, slab, NNODE, NNODE, 384);
  reduce_epi_kernel<2><<<rgrid, 256, 0, stream>>>(slab, 4, buf4, NNODE,
                                                  b2, deg, buf2);
  // A3 = x2 @ W3_top ; B3 = x2 @ W3_bot
  wmma_gemm_part<<<g1, blk, 0, stream>>>(buf4, W3, buf0, NNODE, HID, HID);
  wmma_gemm_part<<<g1, blk, 0, stream>>>(buf4, W3 + HID * HID, buf3, NNODE, HID, HID);
  // x3 = deg*A3 + adj1 @ B3 + b3
  wmma_gemm_part<<<g4, blk, 0, stream>>>(adj1, buf3, slab, NNODE, NNODE, 384);
  reduce_epi_kernel<3><<<rgrid, 256, 0, stream>>>(slab, 4, buf1, NNODE,
                                                  b3, deg, buf0);
  // out = log_softmax(x3, axis=1)
  logsoftmax_kernel<<<NNODE / 8, 256, 0, stream>>>(buf1, out, NNODE);
}